// GNNHypernetwork3_10677288698528
// MI455X (gfx1250) — compile-verified
//
#include <hip/hip_runtime.h>
#include <hip/hip_bf16.h>

#define NN 2048      // nodes per graph
#define MM 64        // input feature dim
#define EE 65536     // edges
#define BB 8         // graphs
#define EPSV 1e-5f

typedef float v2f __attribute__((ext_vector_type(2)));
typedef float v8f __attribute__((ext_vector_type(8)));

// ---------------------------------------------------------------------------
// Degree / GCN norm precompute (adjacency shared across the batch)
// ---------------------------------------------------------------------------
__global__ void k_deg_init(float* __restrict__ deg) {
    int n = blockIdx.x * blockDim.x + threadIdx.x;
    if (n < NN) deg[n] = 1.0f;  // self-loop
}

__global__ void k_deg_count(float* __restrict__ deg, const long long* __restrict__ dst) {
    int e = blockIdx.x * blockDim.x + threadIdx.x;
    if (e < EE) atomicAdd(&deg[(int)dst[e]], 1.0f);
}

__global__ void k_edge_norm(const float* __restrict__ deg,
                            const long long* __restrict__ src,
                            const long long* __restrict__ dst,
                            float* __restrict__ enorm, float* __restrict__ snorm) {
    int i = blockIdx.x * blockDim.x + threadIdx.x;
    if (i < EE) {
        enorm[i] = rsqrtf(deg[(int)src[i]]) * rsqrtf(deg[(int)dst[i]]);
    } else if (i < EE + NN) {
        int n = i - EE;
        float r = rsqrtf(deg[n]);
        snorm[n] = r * r;
    }
}

// ---------------------------------------------------------------------------
// WMMA f32 GEMM: Y[b,n,o] = sum_k X[b,n,k] * W[o,k]   (Y = X @ W^T)
// Block = 128 threads (4 waves), 64x64 block tile, 32x32 per wave
// (2x2 grid of 16x16 WMMA accumulators), K-step 16.
// Data movement: gfx1250 GLOBAL_LOAD_ASYNC_TO_LDS_B128 into double-buffered
// LDS tiles (ASYNCcnt), overlapping DMA with the WMMA section.
// Both A and B tiles are stored row-major ([row][k], row stride 20 floats =
// 80B: 16B-aligned B128 destinations, 20*l16 mod 64 -> conflict-free reads).
// ---------------------------------------------------------------------------
__global__ __launch_bounds__(128)
void k_gemm_wmma(const float* __restrict__ X, const float* __restrict__ W,
                 float* __restrict__ Y, int Di, int Do) {
    __shared__ __align__(16) float As[2][64][20];   // [buf][m][k]
    __shared__ __align__(16) float Bs[2][64][20];   // [buf][o][k]  (W rows)

    const int m0   = blockIdx.x * 64;
    const int n0   = blockIdx.y * 64;
    const int b    = blockIdx.z;
    const int t    = threadIdx.x;
    const int wave = t >> 5;
    const int lane = t & 31;
    const int half = lane >> 4;     // lane 16..31 supplies K+2 slots
    const int l16  = lane & 15;
    const int wm   = (wave >> 1) * 32;
    const int wn   = (wave & 1) * 32;

    const float* Xg = X + (size_t)b * NN * Di;
    float*       Yg = Y + (size_t)b * NN * Do;

    // cooperative load mapping: 64 rows x 16 k per tile, 2 x B128 per thread/array
    const int lr  = t >> 2;        // 0..31 (also lr+32)
    const int lc4 = (t & 3) * 4;   // 0,4,8,12

    const float* gA0 = Xg + (size_t)(m0 + lr)      * Di + lc4;
    const float* gA1 = Xg + (size_t)(m0 + lr + 32) * Di + lc4;
    const float* gB0 = W  + (size_t)(n0 + lr)      * Di + lc4;
    const float* gB1 = W  + (size_t)(n0 + lr + 32) * Di + lc4;

    // issue 4 async B128 global->LDS copies for K-tile at offset kof into buffer bi
    auto issue = [&](int bi, int kof) {
        asm volatile(
            "global_load_async_to_lds_b128 %0, %4, off\n\t"
            "global_load_async_to_lds_b128 %1, %5, off\n\t"
            "global_load_async_to_lds_b128 %2, %6, off\n\t"
            "global_load_async_to_lds_b128 %3, %7, off"
            :
            : "v"((unsigned)(uintptr_t)&As[bi][lr][lc4]),
              "v"((unsigned)(uintptr_t)&As[bi][lr + 32][lc4]),
              "v"((unsigned)(uintptr_t)&Bs[bi][lr][lc4]),
              "v"((unsigned)(uintptr_t)&Bs[bi][lr + 32][lc4]),
              "v"(gA0 + kof), "v"(gA1 + kof), "v"(gB0 + kof), "v"(gB1 + kof)
            : "memory");
    };

    v8f acc00 = {0.f,0.f,0.f,0.f,0.f,0.f,0.f,0.f};
    v8f acc01 = acc00, acc10 = acc00, acc11 = acc00;

    issue(0, 0);
    asm volatile("s_wait_asynccnt 0x0" ::: "memory");
    __syncthreads();

    int buf = 0;
    for (int k0 = 0; k0 < Di; k0 += 16) {
        // start DMA of the next K-tile while this one is consumed by WMMA
        if (k0 + 16 < Di) issue(buf ^ 1, k0 + 16);

#pragma unroll
        for (int kk = 0; kk < 4; ++kk) {
            const int kb = kk * 4 + half * 2;  // lane halves supply K / K+2 pairs
            v2f a0, a1, b0, b1;
            a0.x = As[buf][wm + l16][kb];        a0.y = As[buf][wm + l16][kb + 1];
            a1.x = As[buf][wm + 16 + l16][kb];   a1.y = As[buf][wm + 16 + l16][kb + 1];
            b0.x = Bs[buf][wn + l16][kb];        b0.y = Bs[buf][wn + l16][kb + 1];
            b1.x = Bs[buf][wn + 16 + l16][kb];   b1.y = Bs[buf][wn + 16 + l16][kb + 1];
            acc00 = __builtin_amdgcn_wmma_f32_16x16x4_f32(false, a0, false, b0, (short)0, acc00, false, false);
            acc01 = __builtin_amdgcn_wmma_f32_16x16x4_f32(false, a0, false, b1, (short)0, acc01, false, false);
            acc10 = __builtin_amdgcn_wmma_f32_16x16x4_f32(false, a1, false, b0, (short)0, acc10, false, false);
            acc11 = __builtin_amdgcn_wmma_f32_16x16x4_f32(false, a1, false, b1, (short)0, acc11, false, false);
        }

        // drain this wave's outstanding async DMA, then block-wide barrier:
        // all waves done reading buf, next buffer fully landed in LDS
        asm volatile("s_wait_asynccnt 0x0" ::: "memory");
        __syncthreads();
        buf ^= 1;
    }

    // C/D layout: VGPR r -> lanes 0-15: M=r, N=lane; lanes 16-31: M=r+8, N=lane-16
    const int cm = m0 + wm + half * 8;
    const int cn = n0 + wn + l16;
#pragma unroll
    for (int r = 0; r < 8; ++r) {
        Yg[(size_t)(cm + r) * Do + cn]           = acc00[r];
        Yg[(size_t)(cm + r) * Do + cn + 16]      = acc01[r];
        Yg[(size_t)(cm + 16 + r) * Do + cn]      = acc10[r];
        Yg[(size_t)(cm + 16 + r) * Do + cn + 16] = acc11[r];
    }
}

// ---------------------------------------------------------------------------
// Zero a float buffer
// ---------------------------------------------------------------------------
__global__ void k_zero(float* __restrict__ p, size_t n) {
    size_t i = (size_t)blockIdx.x * blockDim.x + threadIdx.x;
    if (i < n) p[i] = 0.f;
}

// ---------------------------------------------------------------------------
// Edge scatter: agg[dst] += h[src] * norm  (edges + self-loops), per graph
// grid.x = EE+NN, grid.y = B
// ---------------------------------------------------------------------------
__global__ __launch_bounds__(256)
void k_scatter(const float* __restrict__ H, float* __restrict__ AGG,
               const long long* __restrict__ src, const long long* __restrict__ dst,
               const float* __restrict__ enorm, const float* __restrict__ snorm,
               int Do) {
    const int e = blockIdx.x;
    const int b = blockIdx.y;
    int s, d; float w;
    if (e < EE) {
        s = (int)src[e]; d = (int)dst[e]; w = enorm[e];
    } else {
        const int n = e - EE;
        s = n; d = n; w = snorm[n];
    }
    const float* hr = H   + ((size_t)b * NN + s) * Do;
    float*       ar = AGG + ((size_t)b * NN + d) * Do;
    for (int o = threadIdx.x; o < Do; o += blockDim.x)
        atomicAdd(&ar[o], hr[o] * w);
}

// ---------------------------------------------------------------------------
// Fused bias + leakyReLU(0.01) + batchnorm over nodes (training stats, biased var)
// One block per (feature o, graph b). 256 threads x 8 nodes in registers.
// ---------------------------------------------------------------------------
__global__ __launch_bounds__(256)
void k_bias_act_bn(float* __restrict__ H, const float* __restrict__ bias,
                   const float* __restrict__ g, const float* __restrict__ be,
                   int Do) {
    const int o = blockIdx.x;
    const int b = blockIdx.y;
    const int t = threadIdx.x;
    float* col = H + (size_t)b * NN * Do + o;

    float v[8];
    float s = 0.f, s2 = 0.f;
#pragma unroll
    for (int i = 0; i < 8; ++i) {
        float x = col[(size_t)(t + i * 256) * Do] + bias[o];
        x = x > 0.f ? x : 0.01f * x;
        v[i] = x; s += x; s2 += x * x;
    }

    __shared__ float rs[256], rq[256];
    rs[t] = s; rq[t] = s2;
    __syncthreads();
    for (int off = 128; off > 0; off >>= 1) {
        if (t < off) { rs[t] += rs[t + off]; rq[t] += rq[t + off]; }
        __syncthreads();
    }
    __shared__ float bc_scale, bc_shift;
    if (t == 0) {
        const float mu  = rs[0] * (1.0f / NN);
        const float var = rq[0] * (1.0f / NN) - mu * mu;
        const float scl = rsqrtf(var + EPSV) * g[o];
        bc_scale = scl;
        bc_shift = be[o] - mu * scl;
    }
    __syncthreads();
    const float scl = bc_scale, shf = bc_shift;
#pragma unroll
    for (int i = 0; i < 8; ++i)
        col[(size_t)(t + i * 256) * Do] = v[i] * scl + shf;
}

// ---------------------------------------------------------------------------
// Final per-node layernorm over 1024 features -> d_out[b, n*1024 + o]
// ---------------------------------------------------------------------------
__global__ __launch_bounds__(256)
void k_layernorm(const float* __restrict__ H, const float* __restrict__ g,
                 const float* __restrict__ be, float* __restrict__ out) {
    const int n = blockIdx.x;
    const int b = blockIdx.y;
    const int t = threadIdx.x;
    const float* row  = H   + ((size_t)b * NN + n) * 1024;
    float*       orow = out + ((size_t)b * NN + n) * 1024;

    float v[4];
    float s = 0.f, s2 = 0.f;
#pragma unroll
    for (int j = 0; j < 4; ++j) {
        float x = row[t + j * 256];
        v[j] = x; s += x; s2 += x * x;
    }
    __shared__ float rs[256], rq[256];
    rs[t] = s; rq[t] = s2;
    __syncthreads();
    for (int off = 128; off > 0; off >>= 1) {
        if (t < off) { rs[t] += rs[t + off]; rq[t] += rq[t + off]; }
        __syncthreads();
    }
    __shared__ float bc_mu, bc_rstd;
    if (t == 0) {
        const float mu  = rs[0] * (1.0f / 1024.0f);
        const float var = rq[0] * (1.0f / 1024.0f) - mu * mu;
        bc_mu = mu; bc_rstd = rsqrtf(var + EPSV);
    }
    __syncthreads();
    const float mu = bc_mu, rstd = bc_rstd;
#pragma unroll
    for (int j = 0; j < 4; ++j) {
        const int o = t + j * 256;
        orow[o] = (v[j] - mu) * rstd * g[o] + be[o];
    }
}

// ---------------------------------------------------------------------------
// Host launcher
// ---------------------------------------------------------------------------
extern "C" void kernel_launch(void* const* d_in, const int* in_sizes, int n_in,
                              void* d_out, int out_size, void* d_ws, size_t ws_size,
                              hipStream_t stream) {
    (void)in_sizes; (void)n_in; (void)out_size; (void)ws_size;

    const float*     x   = (const float*)d_in[22];         // [B, N, 64]
    const long long* ei  = (const long long*)d_in[23];     // [2, E] int64
    const long long* src = ei;
    const long long* dst = ei + EE;
    const float*     lng = (const float*)d_in[20];
    const float*     lnb = (const float*)d_in[21];
    float*           out = (float*)d_out;

    // Workspace layout (floats): two ping-pong activation buffers + norm scratch
    const size_t BIG = (size_t)BB * NN * 1024;
    float* fws   = (float*)d_ws;
    float* tmp   = fws;              // GEMM output
    float* act   = fws + BIG;        // aggregated / normalized activations
    float* deg   = fws + 2 * BIG;    // [N]
    float* enorm = deg + NN;         // [E]
    float* snorm = enorm + EE;       // [N]

    // --- GCN normalization (shared across batch) ---
    k_deg_init <<<(NN + 255) / 256, 256, 0, stream>>>(deg);
    k_deg_count<<<(EE + 255) / 256, 256, 0, stream>>>(deg, dst);
    k_edge_norm<<<(EE + NN + 255) / 256, 256, 0, stream>>>(deg, src, dst, enorm, snorm);

    struct Layer { const float *w, *b, *g, *be; int Di, Do; };
    const Layer L[5] = {
        { (const float*)d_in[0],  (const float*)d_in[1],  (const float*)d_in[2],  (const float*)d_in[3],    64,  256 },
        { (const float*)d_in[4],  (const float*)d_in[5],  (const float*)d_in[6],  (const float*)d_in[7],   256,  512 },
        { (const float*)d_in[8],  (const float*)d_in[9],  (const float*)d_in[10], (const float*)d_in[11],  512, 1024 },
        { (const float*)d_in[12], (const float*)d_in[13], (const float*)d_in[14], (const float*)d_in[15], 1024, 1024 },
        { (const float*)d_in[16], (const float*)d_in[17], (const float*)d_in[18], (const float*)d_in[19], 1024, 1024 },
    };

    const float* cur = x;
    for (int l = 0; l < 5; ++l) {
        const int Di = L[l].Di, Do = L[l].Do;

        dim3 gg(NN / 64, Do / 64, BB);
        k_gemm_wmma<<<gg, 128, 0, stream>>>(cur, L[l].w, tmp, Di, Do);

        const size_t cnt = (size_t)BB * NN * Do;
        k_zero<<<(unsigned)((cnt + 255) / 256), 256, 0, stream>>>(act, cnt);

        dim3 gs(EE + NN, BB);
        k_scatter<<<gs, 256, 0, stream>>>(tmp, act, src, dst, enorm, snorm, Do);

        dim3 gb(Do, BB);
        k_bias_act_bn<<<gb, 256, 0, stream>>>(act, L[l].b, L[l].g, L[l].be, Do);

        cur = act;
    }

    dim3 gl(NN, BB);
    k_layernorm<<<gl, 256, 0, stream>>>(act, lng, lnb, out);
}